// ConfigurationDecoder_55018531061883
// MI455X (gfx1250) — compile-verified
//
#include <hip/hip_runtime.h>
#include <hip/hip_bf16.h>
#include <math.h>

typedef __attribute__((ext_vector_type(2))) float v2f;
typedef __attribute__((ext_vector_type(8))) float v8f;

#define BDIM   64
#define PANO   36
#define CANDN  16
#define OBJN   16
#define CFGN   16
#define LMKN   8
#define DD     300
#define HH     512
#define EMBN   64
#define ANGN   4
#define FEATN  2052
#define TOPK   3
#define CTXD   2952      // FEATN + 3*DD
#define CL     128       // CFGN*LMKN

// ---------------------------------------------------------------------------
// K0: normalize landmark_object_feature rows -> lofn (B, 128, 300)
// one wave per row
// ---------------------------------------------------------------------------
__global__ void lof_norm_kernel(const float* __restrict__ lof,
                                float* __restrict__ lofn) {
  int wave = threadIdx.x >> 5, lane = threadIdx.x & 31;
  int rid = blockIdx.x * 8 + wave;                 // row over B*CL = 8192
  const float* src = lof + (size_t)rid * DD;
  float ss = 0.f;
  for (int k = lane; k < DD; k += 32) { float v = src[k]; ss += v * v; }
  for (int off = 16; off; off >>= 1) ss += __shfl_down(ss, off, 32);
  ss = __shfl(ss, 0, 32);
  float scale = 1.f / fmaxf(sqrtf(ss), 1e-8f);
  float* dst = lofn + (size_t)rid * DD;
  for (int k = lane; k < DD; k += 32) dst[k] = src[k] * scale;
}

// ---------------------------------------------------------------------------
// K1: cosine sim via fp32 WMMA + max/argmax over O.
// grid = B*I blocks, 256 threads (8 waves). Wave w computes columns w*16..+15.
// A (16x300) = normalized (optionally masked) object feats, staged in LDS.
// B rows = lofn[b] (each output column n corresponds to lofn row n; dot over k).
// outputs: val (B,I,128), amax (B,I,128)
// ---------------------------------------------------------------------------
__global__ void sim_retrieve_kernel(const float* __restrict__ feat,
                                    const float* __restrict__ mask,
                                    const float* __restrict__ lofn,
                                    float* __restrict__ val,
                                    int* __restrict__ amax,
                                    int I) {
  __shared__ float As[16][304];
  int bi = blockIdx.x;
  int b  = bi / I;
  int wave = threadIdx.x >> 5, lane = threadIdx.x & 31;
  const float* A0 = feat + (size_t)bi * OBJN * DD;

  // normalize rows (mask*row / max(||mask*row||, eps)), 8 waves x 2 rows
  for (int o = wave; o < OBJN; o += 8) {
    float mv = mask ? mask[(size_t)bi * OBJN + o] : 1.f;
    const float* row = A0 + (size_t)o * DD;
    float ss = 0.f;
    for (int k = lane; k < DD; k += 32) { float v = row[k] * mv; ss += v * v; }
    for (int off = 16; off; off >>= 1) ss += __shfl_down(ss, off, 32);
    ss = __shfl(ss, 0, 32);
    float scale = mv / fmaxf(sqrtf(ss), 1e-8f);
    for (int k = lane; k < DD; k += 32) As[o][k] = row[k] * scale;
  }
  __syncthreads();

  int grp = lane >> 4;            // half-wave: K sub-offset group
  int lm  = lane & 15;            // M for A-frag, N for B-frag
  const float* Brow = lofn + ((size_t)b * CL + wave * 16 + lm) * DD;
  v8f acc = {};
  for (int k0 = 0; k0 < DD; k0 += 4) {
    int kb = k0 + grp * 2;
    v2f a;  a.x = As[lm][kb];  a.y = As[lm][kb + 1];
    v2f bf; bf.x = Brow[kb];   bf.y = Brow[kb + 1];
    acc = __builtin_amdgcn_wmma_f32_16x16x4_f32(false, a, false, bf,
                                                (short)0, acc, false, false);
  }
  // acc VGPR r holds sim[M = r + 8*grp][N = wave*16+lm]; reduce over M
  float best = acc[0];
  int barg = 8 * grp;
  for (int r = 1; r < 8; ++r) {
    float v = acc[r];
    if (v > best) { best = v; barg = r + 8 * grp; }     // first-max tie-break
  }
  float obest = __shfl_xor(best, 16, 32);
  int   oarg  = __shfl_xor(barg, 16, 32);
  if (obest > best || (obest == best && oarg < barg)) { best = obest; barg = oarg; }
  if (grp == 0) {
    size_t p = (size_t)bi * CL + wave * 16 + lm;
    val[p] = best;
    amax[p] = barg;
  }
}

// ---------------------------------------------------------------------------
// K2: per-batch top-3 of scores = wscore[b,c] * lmask[b,c,l] (stable descending)
// ---------------------------------------------------------------------------
__global__ void topn_kernel(const float* __restrict__ wscore,
                            const float* __restrict__ lmask,
                            int* __restrict__ topn) {
  int b = blockIdx.x * blockDim.x + threadIdx.x;
  if (b >= BDIM) return;
  float s[CL];
  for (int c = 0; c < CFGN; ++c) {
    float w = wscore[b * CFGN + c];
    for (int l = 0; l < LMKN; ++l)
      s[c * LMKN + l] = w * lmask[((size_t)b * CFGN + c) * LMKN + l];
  }
  for (int j = 0; j < TOPK; ++j) {
    int bi = 0; float bv = -INFINITY;
    for (int p = 0; p < CL; ++p)
      if (s[p] > bv) { bv = s[p]; bi = p; }   // strict > == stable (first)
    s[bi] = -INFINITY;
    topn[b * TOPK + j] = bi;
  }
}

// ---------------------------------------------------------------------------
// K3: rank-select within the l-block (stable descending by val), then gather
// gather_feat rows into out (B,I,3*300)
// ---------------------------------------------------------------------------
__global__ void gather_topn_kernel(const float* __restrict__ val,
                                   const int* __restrict__ amax,
                                   const int* __restrict__ topn,
                                   const float* __restrict__ gfeat,
                                   float* __restrict__ out, int I) {
  __shared__ int selObj[TOPK];
  int bi = blockIdx.x;
  int b  = bi / I;
  int tid = threadIdx.x;
  if (tid < TOPK) {
    int p = topn[b * TOPK + tid];
    int c = p >> 3, lp = p & 7;
    const float* v = val + (size_t)bi * CL + c * LMKN;
    int sel = 0;
    for (int l = 0; l < LMKN; ++l) {
      int rank = 0;
      for (int l2 = 0; l2 < LMKN; ++l2) {
        float a = v[l2], bb = v[l];
        rank += (a > bb || (a == bb && l2 < l)) ? 1 : 0;
      }
      if (rank == lp) sel = l;
    }
    selObj[tid] = amax[(size_t)bi * CL + c * LMKN + sel];
  }
  __syncthreads();
  for (int t = tid; t < TOPK * DD; t += blockDim.x) {
    int j = t / DD, d = t - j * DD;
    out[(size_t)bi * (TOPK * DD) + t] =
        gfeat[((size_t)bi * OBJN + selObj[j]) * DD + d];
  }
}

// ---------------------------------------------------------------------------
// Generic WMMA GEMM: C(64 x N) = act( A(64 x K) * W(N x K)^T + bias [+ C] )
// One wave per 16x16 tile; block = 256 (8 n-tiles); grid = (ceil(Ntiles/8), 4).
// K must be a multiple of 4 (true for 512/1024/3016). Ragged N handled by
// clamping W row pointer and zero-masking the B fragment (no EXEC divergence
// around the WMMA).
// ---------------------------------------------------------------------------
__global__ void gemm_nt_wmma_kernel(const float* __restrict__ A, int lda,
                                    const float* __restrict__ W, int ldw,
                                    float* __restrict__ C, int ldc,
                                    int N, int K,
                                    const float* __restrict__ bias,
                                    int accumulate, int act) {
  int wave = threadIdx.x >> 5, lane = threadIdx.x & 31;
  int ntile = blockIdx.x * 8 + wave;
  int m0 = blockIdx.y * 16;
  int n0 = ntile * 16;
  if (n0 >= N) return;                       // uniform per wave
  int grp = lane >> 4, lm = lane & 15;
  int m = m0 + lm;
  int n = n0 + lm;
  bool nv = n < N;
  const float* Arow = A + (size_t)m * lda;
  const float* Wrow = W + (size_t)(nv ? n : (N - 1)) * ldw;
  float wm = nv ? 1.f : 0.f;
  v8f acc = {};
  for (int k0 = 0; k0 < K; k0 += 4) {
    int kb = k0 + grp * 2;
    v2f a;  a.x = Arow[kb];       a.y = Arow[kb + 1];
    v2f bf; bf.x = Wrow[kb] * wm; bf.y = Wrow[kb + 1] * wm;
    acc = __builtin_amdgcn_wmma_f32_16x16x4_f32(false, a, false, bf,
                                                (short)0, acc, false, false);
  }
  if (!nv) return;
  float bc = bias ? bias[n] : 0.f;
  for (int r = 0; r < 8; ++r) {
    int row = m0 + r + 8 * grp;
    size_t off = (size_t)row * ldc + n;
    float v = acc[r] + bc;
    if (accumulate) v += C[off];
    if (act) v = tanhf(v);
    C[off] = v;
  }
}

// ---------------------------------------------------------------------------
// action embedding -> xbuf[b, 0:64]
// ---------------------------------------------------------------------------
__global__ void action_embed_kernel(const float* __restrict__ action,
                                    const float* __restrict__ Wemb,
                                    const float* __restrict__ bemb,
                                    float* __restrict__ xbuf) {
  int b = blockIdx.x, e = threadIdx.x;
  float s = bemb[e];
  for (int k = 0; k < ANGN; ++k) s += action[b * ANGN + k] * Wemb[e * ANGN + k];
  xbuf[(size_t)b * (EMBN + CTXD) + e] = tanhf(s);
}

// ---------------------------------------------------------------------------
// feature attention: logits over 36 panos (feature ++ pano_sim dot tq), softmax
// ---------------------------------------------------------------------------
__global__ void feat_attn_logits_kernel(const float* __restrict__ feature,
                                        const float* __restrict__ psim,
                                        const float* __restrict__ tq,
                                        float* __restrict__ aout) {
  int b = blockIdx.x;
  __shared__ float lg[PANO];
  int wave = threadIdx.x >> 5, lane = threadIdx.x & 31;
  const float* t = tq + (size_t)b * CTXD;
  for (int s = wave; s < PANO; s += 8) {
    const float* f = feature + ((size_t)b * PANO + s) * FEATN;
    const float* p = psim + ((size_t)b * PANO + s) * (TOPK * DD);
    float acc = 0.f;
    for (int d = lane; d < FEATN; d += 32) acc += f[d] * t[d];
    for (int d = lane; d < TOPK * DD; d += 32) acc += p[d] * t[FEATN + d];
    for (int off = 16; off; off >>= 1) acc += __shfl_down(acc, off, 32);
    if (lane == 0) lg[s] = acc;
  }
  __syncthreads();
  if (threadIdx.x == 0) {
    float mx = -INFINITY;
    for (int s = 0; s < PANO; ++s) mx = fmaxf(mx, lg[s]);
    float sum = 0.f;
    for (int s = 0; s < PANO; ++s) { float e = __expf(lg[s] - mx); lg[s] = e; sum += e; }
    float inv = 1.f / sum;
    for (int s = 0; s < PANO; ++s) aout[b * PANO + s] = lg[s] * inv;
  }
}

// weighted context -> xbuf[b, 64 : 64+2952]
__global__ void feat_attn_wsum_kernel(const float* __restrict__ feature,
                                      const float* __restrict__ psim,
                                      const float* __restrict__ a,
                                      float* __restrict__ xbuf) {
  int b = blockIdx.x;
  __shared__ float aw[PANO];
  if (threadIdx.x < PANO) aw[threadIdx.x] = a[b * PANO + threadIdx.x];
  __syncthreads();
  for (int d = threadIdx.x; d < CTXD; d += blockDim.x) {
    float acc = 0.f;
    if (d < FEATN) {
      const float* f = feature + (size_t)b * PANO * FEATN + d;
      for (int s = 0; s < PANO; ++s) acc += aw[s] * f[(size_t)s * FEATN];
    } else {
      const float* p = psim + (size_t)b * PANO * (TOPK * DD) + (d - FEATN);
      for (int s = 0; s < PANO; ++s) acc += aw[s] * p[(size_t)s * (TOPK * DD)];
    }
    xbuf[(size_t)b * (EMBN + CTXD) + EMBN + d] = acc;
  }
}

// ---------------------------------------------------------------------------
// LSTM cell elementwise (gate order i,f,g,o)
// ---------------------------------------------------------------------------
__global__ void lstm_cell_kernel(const float* __restrict__ gates,
                                 const float* __restrict__ c0,
                                 float* __restrict__ h1,
                                 float* __restrict__ c1) {
  int b = blockIdx.x, h = threadIdx.x;
  const float* g = gates + (size_t)b * 4 * HH;
  float ig = g[h], fg = g[HH + h], gg = g[2 * HH + h], og = g[3 * HH + h];
  float si = 1.f / (1.f + __expf(-ig));
  float sf = 1.f / (1.f + __expf(-fg));
  float so = 1.f / (1.f + __expf(-og));
  float cc = sf * c0[(size_t)b * HH + h] + si * tanhf(gg);
  c1[(size_t)b * HH + h] = cc;
  h1[(size_t)b * HH + h] = so * tanhf(cc);
}

// ---------------------------------------------------------------------------
// ctx attention: masked softmax over 16 slots, weighted ctx, build cat=[wctx,h1]
// one wave per batch
// ---------------------------------------------------------------------------
__global__ void ctx_attn_kernel(const float* __restrict__ ctx,
                                const unsigned char* __restrict__ cmask,
                                const float* __restrict__ q,
                                const float* __restrict__ h1,
                                float* __restrict__ attn_out,
                                float* __restrict__ cat) {
  int b = blockIdx.x;
  int lane = threadIdx.x;
  float lg = -INFINITY;
  if (lane < CFGN) {
    const float* c = ctx + ((size_t)b * CFGN + lane) * HH;
    const float* qq = q + (size_t)b * HH;
    float acc = 0.f;
    for (int d = 0; d < HH; ++d) acc += c[d] * qq[d];
    lg = cmask[b * CFGN + lane] ? -INFINITY : acc;
  }
  float mx = lg;
  for (int off = 8; off; off >>= 1) mx = fmaxf(mx, __shfl_xor(mx, off, 32));
  float e = (lane < CFGN) ? __expf(lg - mx) : 0.f;
  float sum = e;
  for (int off = 8; off; off >>= 1) sum += __shfl_xor(sum, off, 32);
  float attn = (lane < CFGN) ? e / sum : 0.f;
  if (lane < CFGN) attn_out[b * CFGN + lane] = attn;
  for (int d = lane; d < HH; d += 32) {
    float acc = 0.f;
    for (int s = 0; s < CFGN; ++s) {
      float as = __shfl(attn, s, 32);
      acc += as * ctx[((size_t)b * CFGN + s) * HH + d];
    }
    cat[(size_t)b * 2 * HH + d] = acc;
    cat[(size_t)b * 2 * HH + HH + d] = h1[(size_t)b * HH + d];
  }
}

// ---------------------------------------------------------------------------
// final candidate logits: one wave per (b,c)
// ---------------------------------------------------------------------------
__global__ void cand_logit_kernel(const float* __restrict__ cand_feat,
                                  const float* __restrict__ csim,
                                  const float* __restrict__ cq,
                                  float* __restrict__ logit) {
  int bc = blockIdx.x;
  int b = bc >> 4;
  int lane = threadIdx.x;
  const float* f = cand_feat + (size_t)bc * FEATN;
  const float* p = csim + (size_t)bc * (TOPK * DD);
  const float* t = cq + (size_t)b * CTXD;
  float acc = 0.f;
  for (int d = lane; d < FEATN; d += 32) acc += f[d] * t[d];
  for (int d = lane; d < TOPK * DD; d += 32) acc += p[d] * t[FEATN + d];
  for (int off = 16; off; off >>= 1) acc += __shfl_down(acc, off, 32);
  if (lane == 0) logit[bc] = acc;
}

// ---------------------------------------------------------------------------
// host-side orchestration
// ---------------------------------------------------------------------------
extern "C" void kernel_launch(void* const* d_in, const int* in_sizes, int n_in,
                              void* d_out, int out_size, void* d_ws, size_t ws_size,
                              hipStream_t stream) {
  (void)in_sizes; (void)n_in; (void)out_size; (void)ws_size;
  const float* action    = (const float*)d_in[0];
  const float* feature   = (const float*)d_in[1];
  const float* cand_feat = (const float*)d_in[2];
  // d_in[3] = h_0 (unused by reference)
  const float* prev_h1   = (const float*)d_in[4];
  const float* c_0       = (const float*)d_in[5];
  const float* ctx       = (const float*)d_in[6];
  const float* s_0       = (const float*)d_in[7];
  const unsigned char* ctx_mask = (const unsigned char*)d_in[8];
  const float* lmask     = (const float*)d_in[9];
  const float* lof       = (const float*)d_in[10];
  const float* cand_obj  = (const float*)d_in[11];
  const float* pano_obj  = (const float*)d_in[12];
  const float* pano_mask = (const float*)d_in[13];
  const float* W_emb     = (const float*)d_in[14];
  const float* b_emb     = (const float*)d_in[15];
  const float* W_feat_in = (const float*)d_in[16];
  const float* W_lstm_ih = (const float*)d_in[17];
  const float* W_lstm_hh = (const float*)d_in[18];
  const float* b_lstm_ih = (const float*)d_in[19];
  const float* b_lstm_hh = (const float*)d_in[20];
  const float* W_attn_in = (const float*)d_in[21];
  const float* W_attn_out= (const float*)d_in[22];
  const float* W_cand_in = (const float*)d_in[23];

  float* out = (float*)d_out;
  const size_t H1_OFF = 0;
  const size_t C1_OFF = (size_t)BDIM * HH;          // 32768
  const size_t LG_OFF = C1_OFF + (size_t)BDIM * HH; // 65536
  const size_t HT_OFF = LG_OFF + (size_t)BDIM * CANDN; // 66560
  const size_t CA_OFF = HT_OFF + (size_t)BDIM * HH; // 99328

  float* wsf = (float*)d_ws;
  // workspace layout (float offsets)
  float* LOFN = wsf + 0;                        // 64*128*300
  float* PVAL = LOFN + (size_t)BDIM * CL * DD;  // 64*36*128
  int*   PIDX = (int*)(PVAL + (size_t)BDIM * PANO * CL);
  float* PSIM = (float*)(PIDX + (size_t)BDIM * PANO * CL);       // 64*36*900
  float* CVAL = PSIM + (size_t)BDIM * PANO * (TOPK * DD);        // 64*16*128
  int*   CIDX = (int*)(CVAL + (size_t)BDIM * CANDN * CL);
  float* CSIM = (float*)(CIDX + (size_t)BDIM * CANDN * CL);      // 64*16*900
  int*   TOPP = (int*)(CSIM + (size_t)BDIM * CANDN * (TOPK * DD));
  int*   TOPC = TOPP + BDIM * TOPK;
  float* TQ   = (float*)(TOPC + BDIM * TOPK);   // 64*2952
  float* AW   = TQ + (size_t)BDIM * CTXD;       // 64*36
  float* XBUF = AW + (size_t)BDIM * PANO;       // 64*3016
  float* GATES= XBUF + (size_t)BDIM * (EMBN + CTXD); // 64*2048
  float* QATT = GATES + (size_t)BDIM * 4 * HH;  // 64*512
  float* CAT  = QATT + (size_t)BDIM * HH;       // 64*1024
  float* CQ   = CAT + (size_t)BDIM * 2 * HH;    // 64*2952

  // 0) normalize landmark object features
  lof_norm_kernel<<<(BDIM * CL) / 8, 256, 0, stream>>>(lof, LOFN);

  // 1) pano retrieve: sim (WMMA) -> topn -> gather
  sim_retrieve_kernel<<<BDIM * PANO, 256, 0, stream>>>(pano_obj, pano_mask, LOFN,
                                                       PVAL, PIDX, PANO);
  topn_kernel<<<2, 32, 0, stream>>>(s_0, lmask, TOPP);
  gather_topn_kernel<<<BDIM * PANO, 256, 0, stream>>>(PVAL, PIDX, TOPP, pano_obj,
                                                      PSIM, PANO);

  // 2) tq = prev_h1 @ W_feat_in^T  (64 x 2952), WMMA, ragged N
  {
    int ntiles = (CTXD + 15) / 16;                // 185
    dim3 g((ntiles + 7) / 8, 4);
    gemm_nt_wmma_kernel<<<g, 256, 0, stream>>>(prev_h1, HH, W_feat_in, HH,
                                               TQ, CTXD, CTXD, HH,
                                               nullptr, 0, 0);
  }

  // 3) feature soft-attention
  feat_attn_logits_kernel<<<BDIM, 256, 0, stream>>>(feature, PSIM, TQ, AW);
  feat_attn_wsum_kernel<<<BDIM, 256, 0, stream>>>(feature, PSIM, AW, XBUF);
  action_embed_kernel<<<BDIM, EMBN, 0, stream>>>(action, W_emb, b_emb, XBUF);

  // 4) LSTM gates = x @ W_ih^T + b_ih + prev_h1 @ W_hh^T + b_hh  (64 x 2048)
  {
    dim3 g((2048 / 16 + 7) / 8, 4);               // (16,4)
    gemm_nt_wmma_kernel<<<g, 256, 0, stream>>>(XBUF, EMBN + CTXD,
                                               W_lstm_ih, EMBN + CTXD,
                                               GATES, 4 * HH, 4 * HH, EMBN + CTXD,
                                               b_lstm_ih, 0, 0);
    gemm_nt_wmma_kernel<<<g, 256, 0, stream>>>(prev_h1, HH, W_lstm_hh, HH,
                                               GATES, 4 * HH, 4 * HH, HH,
                                               b_lstm_hh, 1, 0);
  }
  lstm_cell_kernel<<<BDIM, HH, 0, stream>>>(GATES, c_0, out + H1_OFF, out + C1_OFF);

  // 5) ctx attention: q = h1 @ W_attn_in^T ; softmax ; h_tilde
  {
    dim3 g((HH / 16 + 7) / 8, 4);                 // (4,4)
    gemm_nt_wmma_kernel<<<g, 256, 0, stream>>>(out + H1_OFF, HH, W_attn_in, HH,
                                               QATT, HH, HH, HH, nullptr, 0, 0);
  }
  ctx_attn_kernel<<<BDIM, 32, 0, stream>>>(ctx, ctx_mask, QATT, out + H1_OFF,
                                           out + CA_OFF, CAT);
  {
    dim3 g((HH / 16 + 7) / 8, 4);
    gemm_nt_wmma_kernel<<<g, 256, 0, stream>>>(CAT, 2 * HH, W_attn_out, 2 * HH,
                                               out + HT_OFF, HH, HH, 2 * HH,
                                               nullptr, 0, 1 /*tanh*/);
  }

  // 6) candidate retrieve (driven by ctx_attn)
  sim_retrieve_kernel<<<BDIM * CANDN, 256, 0, stream>>>(cand_obj, nullptr, LOFN,
                                                        CVAL, CIDX, CANDN);
  topn_kernel<<<2, 32, 0, stream>>>(out + CA_OFF, lmask, TOPC);
  gather_topn_kernel<<<BDIM * CANDN, 256, 0, stream>>>(CVAL, CIDX, TOPC, cand_obj,
                                                       CSIM, CANDN);

  // 7) cq = h_tilde @ W_cand_in^T (64 x 2952) ; final logits
  {
    int ntiles = (CTXD + 15) / 16;
    dim3 g((ntiles + 7) / 8, 4);
    gemm_nt_wmma_kernel<<<g, 256, 0, stream>>>(out + HT_OFF, HH, W_cand_in, HH,
                                               CQ, CTXD, CTXD, HH, nullptr, 0, 0);
  }
  cand_logit_kernel<<<BDIM * CANDN, 32, 0, stream>>>(cand_feat, CSIM, CQ,
                                                     out + LG_OFF);
}